// ClusteringLoss_35373350649879
// MI455X (gfx1250) — compile-verified
//
#include <hip/hip_runtime.h>

#define N_PTS   500000
#define D_DIM   128
#define K_CL    64
#define NITER   10
#define N_WTILE (N_PTS / 16)     // 31250 wave tiles, exact
#define TPB     256
#define WPB     (TPB / 32)       // 8 waves per block
#define ASSIGN_BLOCKS 2048

typedef __attribute__((ext_vector_type(2))) float v2f;
typedef __attribute__((ext_vector_type(8))) float v8f;

// workspace float layout
#define WS_C      0        // 64*128 centroids (row-major [n][k])
#define WS_C2     8192     // 64 squared norms
#define WS_SUMS   8256     // 64*128 cluster sums
#define WS_CNT    16448    // 64 counts (float)
#define WS_SCAL   16512    // [0]=sum_x2, [1]=sum_best

// ---------------- init: strided centroid seed, c2, zero accumulators ----------
__global__ __launch_bounds__(128) void init_kernel(const float* __restrict__ Z,
        float* __restrict__ c, float* __restrict__ c2,
        float* __restrict__ sums, float* __restrict__ cnt, float* __restrict__ scal)
{
    __shared__ float red[128];
    const int k = blockIdx.x;
    const int d = threadIdx.x;
    const long src = (long)k * (N_PTS / K_CL);
    float v = Z[src * D_DIM + d];
    c[k * D_DIM + d] = v;
    sums[k * D_DIM + d] = 0.0f;
    red[d] = v * v;
    __syncthreads();
    for (int s = 64; s > 0; s >>= 1) {
        if (d < s) red[d] += red[d + s];
        __syncthreads();
    }
    if (d == 0) { c2[k] = red[0]; cnt[k] = 0.0f; }
    if (k == 0 && d < 2) scal[d] = 0.0f;
}

// ---------------- sum of squares of all of Z (for the loss) -------------------
__global__ __launch_bounds__(256) void x2_kernel(const float* __restrict__ Z,
                                                 float* __restrict__ sx2)
{
    const float4* z4 = (const float4*)Z;
    const size_t n4 = (size_t)N_PTS * D_DIM / 4;
    size_t i = (size_t)blockIdx.x * blockDim.x + threadIdx.x;
    const size_t stride = (size_t)gridDim.x * blockDim.x;
    float s = 0.0f;
    for (; i < n4; i += stride) {
        float4 v = z4[i];
        s += v.x * v.x + v.y * v.y + v.z * v.z + v.w * v.w;
    }
#pragma unroll
    for (int o = 16; o > 0; o >>= 1) s += __shfl_xor(s, o, 32);
    if ((threadIdx.x & 31) == 0) unsafeAtomicAdd(sx2, s);
}

// ---------------- fused assign (+accumulate or +loss/labels) ------------------
// mode 0: argmin assignment, accumulate cluster sums/counts
// mode 1: argmin assignment, write labels, accumulate sum of min(c2 - 2*dot)
__global__ __launch_bounds__(256, 2) void assign_kernel(const float* __restrict__ Z,
        const float* __restrict__ c, const float* __restrict__ c2g,
        float* __restrict__ sums_g, float* __restrict__ cnt_g,
        float* __restrict__ sum_best, float* __restrict__ out_cl, const int mode)
{
    // cs_p[kp][n] = { c[n][2kp], c[n][2kp+1] }  (k-pair major, no pad)
    __shared__ v2f   cs_p[(D_DIM / 2) * K_CL];   // 32 KB
    __shared__ float sums_l[K_CL * D_DIM];       // 32 KB per-block cluster sums

    const int tid  = threadIdx.x;
    const int lane = tid & 31;
    const int wv   = tid >> 5;
    const int m16  = lane & 15;
    const int pair = lane >> 4;

    {   // stage centroids as k-pair-major float2s (coalesced global, LDS writes 512 B apart per lane)
        const v2f* c2p = (const v2f*)c;          // [n][kp]
        for (int idx = tid; idx < (D_DIM / 2) * K_CL; idx += TPB) {
            const int n  = idx >> 6;             // cluster
            const int kp = idx & 63;             // k-pair
            cs_p[kp * K_CL + n] = c2p[n * (D_DIM / 2) + kp];
        }
        for (int idx = tid; idx < K_CL * D_DIM; idx += TPB) sums_l[idx] = 0.0f;
    }
    __syncthreads();

    const float c2v0 = c2g[ 0 + m16];
    const float c2v1 = c2g[16 + m16];
    const float c2v2 = c2g[32 + m16];
    const float c2v3 = c2g[48 + m16];

    // per-lane B-fragment base pointers: row (kp0 + pair), column n = nt*16+m16
    const v2f* bp0 = cs_p + pair * K_CL + ( 0 + m16);
    const v2f* bp1 = cs_p + pair * K_CL + (16 + m16);
    const v2f* bp2 = cs_p + pair * K_CL + (32 + m16);
    const v2f* bp3 = cs_p + pair * K_CL + (48 + m16);

    float lossAcc = 0.0f;

    for (int wt = blockIdx.x * WPB + wv; wt < N_WTILE; wt += gridDim.x * WPB) {
        const int rowBase = wt * 16;
        const float* aptr = Z + (size_t)(rowBase + m16) * D_DIM + 2 * pair;

        v8f acc0 = {0.f,0.f,0.f,0.f,0.f,0.f,0.f,0.f};
        v8f acc1 = acc0, acc2 = acc0, acc3 = acc0;

#pragma unroll 4
        for (int k0 = 0; k0 < D_DIM; k0 += 4) {
            v2f a  = *(const v2f*)(aptr + k0);       // A 16x4 f32 fragment
            const int kofs = (k0 / 2) * K_CL;        // folds into DS immediate offset
            v2f b0 = bp0[kofs];
            v2f b1 = bp1[kofs];
            v2f b2 = bp2[kofs];
            v2f b3 = bp3[kofs];
            acc0 = __builtin_amdgcn_wmma_f32_16x16x4_f32(false, a, false, b0, (short)0, acc0, false, false);
            acc1 = __builtin_amdgcn_wmma_f32_16x16x4_f32(false, a, false, b1, (short)0, acc1, false, false);
            acc2 = __builtin_amdgcn_wmma_f32_16x16x4_f32(false, a, false, b2, (short)0, acc2, false, false);
            acc3 = __builtin_amdgcn_wmma_f32_16x16x4_f32(false, a, false, b3, (short)0, acc3, false, false);
        }

        // score = c2[n] - 2*dot ; per-lane argmin over the 4 n-tiles
        float bv[8]; int bi[8];
#pragma unroll
        for (int r = 0; r < 8; ++r) {
            float s0 = c2v0 - 2.0f * acc0[r];
            float s1 = c2v1 - 2.0f * acc1[r];
            float s2 = c2v2 - 2.0f * acc2[r];
            float s3 = c2v3 - 2.0f * acc3[r];
            float v = s0; int i = m16;
            if (s1 < v) { v = s1; i = 16 + m16; }
            if (s2 < v) { v = s2; i = 32 + m16; }
            if (s3 < v) { v = s3; i = 48 + m16; }
            bv[r] = v; bi[r] = i;
        }
        // reduce across the 16-lane half (xor masks 1,2,4,8 stay inside half)
#pragma unroll
        for (int r = 0; r < 8; ++r) {
#pragma unroll
            for (int s = 1; s < 16; s <<= 1) {
                float ov = __shfl_xor(bv[r], s, 32);
                int   oi = __shfl_xor(bi[r], s, 32);
                if (ov < bv[r] || (ov == bv[r] && oi < bi[r])) { bv[r] = ov; bi[r] = oi; }
            }
        }

        int myCl = 0;
#pragma unroll
        for (int mm = 0; mm < 16; ++mm) {
            const int srcLane = (mm < 8) ? 0 : 16;       // rows 0-7 live in lanes 0-15
            const int clm = __shfl(bi[mm & 7], srcLane, 32);
            if (lane == mm) myCl = clm;
            if (mode == 0) {
                const float* xrow = Z + (size_t)(rowBase + mm) * D_DIM;
#pragma unroll
                for (int j = 0; j < 4; ++j)
                    atomicAdd(&sums_l[clm * D_DIM + lane + 32 * j], xrow[lane + 32 * j]);
            }
        }
        if (mode == 0) {
            if (lane < 16) unsafeAtomicAdd(&cnt_g[myCl], 1.0f);
        } else {
            if (lane < 16) out_cl[rowBase + lane] = (float)myCl;
            float ls = bv[0] + bv[1] + bv[2] + bv[3] + bv[4] + bv[5] + bv[6] + bv[7];
            if ((lane & 15) == 0) lossAcc += ls;   // lane0: rows 0-7, lane16: rows 8-15
        }
    }

    if (mode == 0) {
        __syncthreads();
        for (int idx = tid; idx < K_CL * D_DIM; idx += TPB) {
            float v = sums_l[idx];
            if (v != 0.0f) unsafeAtomicAdd(&sums_g[idx], v);
        }
    } else {
        if ((lane & 15) == 0) unsafeAtomicAdd(sum_best, lossAcc);
    }
}

// ---------------- centroid update: c = sums/cnt (0 if empty), c2, re-zero -----
__global__ __launch_bounds__(256) void update_kernel(float* __restrict__ c,
        float* __restrict__ c2, float* __restrict__ sums, float* __restrict__ cnt)
{
    const int t = threadIdx.x;
    for (int idx = t; idx < K_CL * D_DIM; idx += 256) {
        const int n = idx >> 7;
        const float ct = cnt[n];
        c[idx] = (ct > 0.0f) ? sums[idx] / ct : 0.0f;
        sums[idx] = 0.0f;
    }
    __syncthreads();
    if (t < K_CL) {
        float s = 0.0f;
        for (int k = 0; k < D_DIM; ++k) { float v = c[t * D_DIM + k]; s += v * v; }
        c2[t] = s;
        cnt[t] = 0.0f;
    }
}

__global__ void finalize_kernel(const float* __restrict__ scal, float* __restrict__ out)
{
    if (threadIdx.x == 0) out[0] = (scal[0] + scal[1]) / (float)N_PTS;
}

extern "C" void kernel_launch(void* const* d_in, const int* in_sizes, int n_in,
                              void* d_out, int out_size, void* d_ws, size_t ws_size,
                              hipStream_t stream)
{
    const float* Z = (const float*)d_in[0];
    float* out  = (float*)d_out;
    float* ws   = (float*)d_ws;
    float* c    = ws + WS_C;
    float* c2   = ws + WS_C2;
    float* sums = ws + WS_SUMS;
    float* cnt  = ws + WS_CNT;
    float* scal = ws + WS_SCAL;   // [0]=sum_x2, [1]=sum_best

    init_kernel<<<K_CL, 128, 0, stream>>>(Z, c, c2, sums, cnt, scal);
    x2_kernel<<<2048, 256, 0, stream>>>(Z, &scal[0]);
    for (int it = 0; it < NITER; ++it) {
        assign_kernel<<<ASSIGN_BLOCKS, TPB, 0, stream>>>(Z, c, c2, sums, cnt,
                                                         &scal[1], out + 1, 0);
        update_kernel<<<1, 256, 0, stream>>>(c, c2, sums, cnt);
    }
    assign_kernel<<<ASSIGN_BLOCKS, TPB, 0, stream>>>(Z, c, c2, sums, cnt,
                                                     &scal[1], out + 1, 1);
    finalize_kernel<<<1, 1, 0, stream>>>(scal, out);
}